// SOM_90761248899729
// MI455X (gfx1250) — compile-verified
//
#include <hip/hip_runtime.h>

// SOM update for MI455X (gfx1250, wave32).
// Pass 1: BMU search via V_WMMA_F32_16X16X4_F32 self-product diagonal.
// Pass 2: memory-bound fused neighbourhood update (f32, float4 traffic).

typedef __attribute__((ext_vector_type(2))) float v2f;
typedef __attribute__((ext_vector_type(8))) float v8f;

#define SOM_M   256
#define SOM_N   256
#define SOM_DIM 512
#define SOM_ROWS (SOM_M * SOM_N)

__global__ void som_init_ws(unsigned long long* __restrict__ ws) {
    if (threadIdx.x == 0 && blockIdx.x == 0) ws[0] = ~0ull;
}

// One wave handles a 16-row tile. A-matrix layout for 16x4 f32:
//   lane n (n<16), VGPR v   -> E[n][v]       (K = 0,1)
//   lane 16+n,     VGPR v   -> E[n][v + 2]   (K = 2,3)
// B-matrix (4x16) wants exactly the same values in the same slots, so we feed
// the identical 2-VGPR error tile as both A and B: D = E * E^T; diag = d2.
__global__ __launch_bounds__(256) void som_bmu_wmma(
        const float* __restrict__ x,
        const float* __restrict__ W,
        unsigned long long* __restrict__ ws) {
    __shared__ float xs[SOM_DIM];
    for (int i = threadIdx.x; i < SOM_DIM; i += blockDim.x) xs[i] = x[i];
    __syncthreads();

    const int lane       = threadIdx.x & 31;
    const int waveId     = blockIdx.x * 8 + (threadIdx.x >> 5);
    const int rowInTile  = (lane < 16) ? lane : (lane - 16);
    const int row        = waveId * 16 + rowInTile;
    const int koff       = (lane < 16) ? 0 : 2;

    const float* __restrict__ wrow = W + (size_t)row * SOM_DIM + koff;
    const float* __restrict__ xrow = xs + koff;

    v8f acc = {};
#pragma unroll 8
    for (int c = 0; c < SOM_DIM / 4; ++c) {
        v2f wv = *(const v2f*)(wrow + 4 * c);
        v2f xv;
        xv.x = xrow[4 * c + 0];
        xv.y = xrow[4 * c + 1];
        v2f e = xv - wv;
        acc = __builtin_amdgcn_wmma_f32_16x16x4_f32(
            /*neg_a=*/false, e, /*neg_b=*/false, e,
            /*c_mod=*/(short)0, acc, /*reuse_a=*/false, /*reuse_b=*/false);
    }

    // Diagonal extraction: D(M=i,N=i). C/D layout: VGPR v, lanes 0-15 -> M=v,
    // lanes 16-31 -> M=v+8; N = lane%16. So row i<8 lives in lane i, acc[i];
    // row i>=8 lives in lane 16+i, acc[i-8].
    float d2 = 0.0f;
    int   myrow = 0;
    bool  valid = false;
    if (lane < 8)        { d2 = acc[lane];      myrow = waveId * 16 + lane;            valid = true; }
    else if (lane >= 24) { d2 = acc[lane - 24]; myrow = waveId * 16 + 8 + (lane - 24); valid = true; }

    unsigned long long key = valid
        ? (((unsigned long long)__float_as_uint(d2) << 32) | (unsigned)myrow)
        : ~0ull;

    // wave32 butterfly min (packed key preserves argmin tie-break to lowest idx)
    for (int off = 16; off > 0; off >>= 1) {
        unsigned lo  = (unsigned)key;
        unsigned hi  = (unsigned)(key >> 32);
        unsigned olo = __shfl_xor(lo, off, 32);
        unsigned ohi = __shfl_xor(hi, off, 32);
        unsigned long long o = ((unsigned long long)ohi << 32) | olo;
        if (o < key) key = o;
    }
    if (lane == 0) atomicMin(ws, key);
}

// One 128-thread block per row: 512 floats = one float4 per thread.
__global__ __launch_bounds__(128) void som_update(
        const float* __restrict__ x,
        const float* __restrict__ W,
        const int* __restrict__ itp,
        const unsigned long long* __restrict__ ws,
        float* __restrict__ out) {
    const int row = blockIdx.x;
    const int tid = threadIdx.x;

    const int   bmu       = (int)(unsigned)(ws[0] & 0xffffffffull);
    const float itf       = (float)itp[0];
    const float lr_decay  = 1.0f - itf * 0.01f;           // it / EPOCHS(100)
    const float alpha_op  = 0.3f   * lr_decay;            // ALPHA
    const float radius_op = 128.0f * lr_decay;            // RADIUS

    const float bi = (float)(bmu / SOM_M);
    const float bj = (float)(bmu % SOM_N);
    const float ri = (float)(row / SOM_M);
    const float rj = (float)(row % SOM_N);
    const float di = bi - ri;
    const float dj = bj - rj;
    const float bmu_d2 = di * di + dj * dj;
    const float lr = alpha_op * expf(-bmu_d2 / (2.0f * radius_op * radius_op));

    const size_t base = (size_t)row * SOM_DIM + (size_t)tid * 4;
    float4 wv = *(const float4*)(W + base);
    float4 xv = *(const float4*)(x + (size_t)tid * 4);
    float4 o;
    o.x = wv.x + lr * (xv.x - wv.x);
    o.y = wv.y + lr * (xv.y - wv.y);
    o.z = wv.z + lr * (xv.z - wv.z);
    o.w = wv.w + lr * (xv.w - wv.w);
    *(float4*)(out + base) = o;
}

extern "C" void kernel_launch(void* const* d_in, const int* in_sizes, int n_in,
                              void* d_out, int out_size, void* d_ws, size_t ws_size,
                              hipStream_t stream) {
    const float* x  = (const float*)d_in[0];
    const float* W  = (const float*)d_in[1];
    const int*   it = (const int*)d_in[2];
    float* out = (float*)d_out;
    unsigned long long* ws = (unsigned long long*)d_ws;

    som_init_ws<<<1, 1, 0, stream>>>(ws);
    // 65536 rows / 16 rows-per-wave = 4096 waves; 8 waves per 256-thread block.
    som_bmu_wmma<<<512, 256, 0, stream>>>(x, W, ws);
    som_update<<<SOM_ROWS, 128, 0, stream>>>(x, W, it, ws, out);
}